// NearestUpsampleBlock_24790551232564
// MI455X (gfx1250) — compile-verified
//
#include <hip/hip_runtime.h>

// NearestUpsample gather: out[i, :] = (idx[i] < N) ? x[idx[i], :] : 0
// D = 128 floats = 512 bytes per row = exactly one wave32 * float4.
// Branch-free clamped gather + select so all 8 B128 gathers per wave are in
// flight simultaneously; wave-uniform (readfirstlane) row bases give
// SADDR-form loads/stores and scalar index fetches.

typedef __attribute__((ext_vector_type(4))) float v4f;
typedef __attribute__((ext_vector_type(4))) int   v4i;

#define D_DIM            128
#define WAVE_SIZE        32
#define ROWS_PER_WAVE    8
#define WAVES_PER_BLOCK  8
#define BLOCK_THREADS    (WAVE_SIZE * WAVES_PER_BLOCK)   // 256 threads = 8 waves
#define ROWS_PER_BLOCK   (ROWS_PER_WAVE * WAVES_PER_BLOCK) // 64 rows / block

__global__ __launch_bounds__(BLOCK_THREADS)
void nearest_upsample_gather(const float* __restrict__ x,
                             const int*   __restrict__ idx,
                             float*       __restrict__ out,
                             int n_coarse, int m)
{
    const int lane = threadIdx.x & (WAVE_SIZE - 1);
    // Force wave-uniformity so row bases live in SGPRs (SADDR-form VMEM).
    const int wave = __builtin_amdgcn_readfirstlane((int)(threadIdx.x >> 5));
    const int row0 = blockIdx.x * ROWS_PER_BLOCK + wave * ROWS_PER_WAVE;
    const int col  = lane * 4;   // 4 floats (16B) per lane -> 512B per wave

    const unsigned un = (unsigned)n_coarse;

    if (row0 + ROWS_PER_WAVE <= m) {
        // ---- Fast path (all rows of m=800000 when grid divides evenly) ----
        // 8 indices for this wave from a uniform address (scalar-load friendly).
        const v4i s0 = *(const v4i*)(idx + row0);
        const v4i s1 = *(const v4i*)(idx + row0 + 4);
        const int s[ROWS_PER_WAVE] = { s0.x, s0.y, s0.z, s0.w,
                                       s1.x, s1.y, s1.z, s1.w };

        // Phase 1: unconditional clamped gathers -- 8 independent
        // global_load_b128 in flight per wave, single loadcnt wait at use.
        v4f v[ROWS_PER_WAVE];
#pragma unroll
        for (int r = 0; r < ROWS_PER_WAVE; ++r) {
            const int c = ((unsigned)s[r] < un) ? s[r] : 0;  // in-bounds clamp
            v[r] = *(const v4f*)(x + (size_t)c * D_DIM + col);
        }

        // Phase 2: select zeros for the shadow row (s == N) and stream out
        // with non-temporal stores so the 410MB output stream does not evict
        // the L2-resident 102MB gather table.
        const v4f zero = {0.0f, 0.0f, 0.0f, 0.0f};
#pragma unroll
        for (int r = 0; r < ROWS_PER_WAVE; ++r) {
            const v4f val = ((unsigned)s[r] < un) ? v[r] : zero;
            __builtin_nontemporal_store(
                val, (v4f*)(out + (size_t)(row0 + r) * D_DIM + col));
        }
    } else {
        // ---- Tail path (partial wave batch) ----
#pragma unroll 1
        for (int r = 0; r < ROWS_PER_WAVE; ++r) {
            const int row = row0 + r;
            if (row >= m) break;
            const int s = idx[row];
            v4f val = {0.0f, 0.0f, 0.0f, 0.0f};
            if ((unsigned)s < un) {
                val = *(const v4f*)(x + (size_t)s * D_DIM + col);
            }
            __builtin_nontemporal_store(
                val, (v4f*)(out + (size_t)row * D_DIM + col));
        }
    }
}

extern "C" void kernel_launch(void* const* d_in, const int* in_sizes, int n_in,
                              void* d_out, int out_size, void* d_ws, size_t ws_size,
                              hipStream_t stream)
{
    const float* x   = (const float*)d_in[0];   // [N, 128] float32
    const int*   idx = (const int*)d_in[1];     // [M] indices in [0, N]
    float*       out = (float*)d_out;           // [M, 128] float32

    const int n_coarse = in_sizes[0] / D_DIM;   // 200000
    const int m        = in_sizes[1];           // 800000

    const int grid = (m + ROWS_PER_BLOCK - 1) / ROWS_PER_BLOCK;  // 12500 blocks
    nearest_upsample_gather<<<grid, BLOCK_THREADS, 0, stream>>>(x, idx, out, n_coarse, m);
}